// Attention_39659728011622
// MI455X (gfx1250) — compile-verified
//
#include <hip/hip_runtime.h>
#include <math.h>

// Problem dims (fixed by the reference)
#define B_    4
#define TDEC  512
#define TENC  1024
#define H_    128

typedef __attribute__((ext_vector_type(2))) float v2f;
typedef __attribute__((ext_vector_type(8))) float v8f;
typedef __attribute__((__vector_size__(4 * sizeof(int)))) int i4;

#if __has_builtin(__builtin_amdgcn_tanhf)
#define TANH_(x) __builtin_amdgcn_tanhf(x)
#else
#define TANH_(x) tanhf(x)
#endif

// gfx1250 async memory->LDS path (ASYNCcnt-tracked, no VGPR staging)
#if defined(__gfx1250__) && __has_builtin(__builtin_amdgcn_global_load_async_to_lds_b128)
#define USE_ASYNC_LDS 1
#else
#define USE_ASYNC_LDS 0
#endif

// Builtin wants typed vector pointers: (int4 AS1*, int4 AS3*, imm, imm)
#define GLOBAL_I4(p) ((__attribute__((address_space(1))) i4*)(p))
#define LDS_I4(p)    ((__attribute__((address_space(3))) i4*)(p))

__device__ __forceinline__ void wait_async_zero() {
#if __has_builtin(__builtin_amdgcn_s_wait_asynccnt)
  __builtin_amdgcn_s_wait_asynccnt(0);
#else
  asm volatile("s_wait_asynccnt 0x0" ::: "memory");
#endif
}

// ---------------------------------------------------------------------------
// Kernel A: dec_t = dec @ W^T + b   via V_WMMA_F32_16X16X4_F32 (fp32 WMMA)
// One wave per 16x16 output tile; K=128 consumed in 32 steps of K=4.
// A layout (16x4 f32): lanes 0-15 -> K=0,1 ; lanes 16-31 -> K=2,3 (2 VGPRs)
// B layout (4x16 f32): same structure with N = lane%16
// C/D layout: VGPR r -> M = r (+8 for lanes>=16), N = lane%16
// ---------------------------------------------------------------------------
__global__ void __launch_bounds__(32) dec_transform_wmma(
    const float* __restrict__ dec,   // (B*TDEC, H)
    const float* __restrict__ Ww,    // (H, H) row-major: Ww[k_out][h]
    const float* __restrict__ Wb,    // (H)
    float*       __restrict__ dec_t) // (B*TDEC, H)
{
  const int I    = blockIdx.x;           // M tile (B*TDEC/16 = 128)
  const int J    = blockIdx.y;           // N tile (H/16 = 8)
  const int lane = threadIdx.x;          // wave32
  const int l16  = lane & 15;
  const int ks   = (lane >> 4) * 2;      // K sub-offset: 0 or 2
  const int rowA = I * 16 + l16;         // A-matrix row (M)
  const int colB = J * 16 + l16;         // B-matrix col (N = output k)

  const float bias = Wb[colB];
  v8f c;
  #pragma unroll
  for (int r = 0; r < 8; ++r) c[r] = bias;   // fold +b into the accumulator

  const float* aptr = dec + (size_t)rowA * H_;
  const float* bptr = Ww  + (size_t)colB * H_;
  #pragma unroll 4
  for (int kk = 0; kk < H_; kk += 4) {
    v2f a, b;
    a.x = aptr[kk + ks]; a.y = aptr[kk + ks + 1];
    b.x = bptr[kk + ks]; b.y = bptr[kk + ks + 1];
    c = __builtin_amdgcn_wmma_f32_16x16x4_f32(false, a, false, b,
                                              (short)0, c, false, false);
  }

  const int mbase = I * 16 + ((lane >> 4) << 3);
  #pragma unroll
  for (int r = 0; r < 8; ++r)
    dec_t[(size_t)(mbase + r) * H_ + colB] = c[r];
}

// ---------------------------------------------------------------------------
// Kernel B (fused): scores -> softmax -> context, one block per (b, 16 d-rows)
// Phase 1: tanh-score (VALU/TRANS bound). enc streamed into double-buffered
//          LDS tiles with GLOBAL_LOAD_ASYNC_TO_LDS_B128: DMA for tile et+1
//          runs underneath the tanh loop for tile et (ASYNCcnt pipelining).
// Phase 2: wave32 softmax (each wave reduces exactly the rows it wrote)
// Phase 3: context = attn @ enc on V_WMMA_F32_16X16X4_F32, 1 h-tile per wave
// ---------------------------------------------------------------------------
#define ETILE 32                       // e-rows per LDS tile
#define NTILE (TENC / ETILE)           // 32 tiles

__global__ void __launch_bounds__(256) attn_fused(
    const float* __restrict__ enc,    // (B, TENC, H)
    const float* __restrict__ Vw,     // (H)
    const float* __restrict__ Vb,     // scalar
    const float* __restrict__ dec_t,  // ws: (B*TDEC, H)
    float*       __restrict__ scores, // ws: (B*TDEC, TENC)
    float*       __restrict__ out)    // (B, TDEC, H)
{
  __shared__ float enc_s[2][ETILE][132];  // double buffer, padded rows
  __shared__ float dect_s[16][132];
  __shared__ float vw_s[H_];

  const int t  = threadIdx.x;         // 256 threads = 8 waves
  const int b  = blockIdx.y;
  const int d0 = blockIdx.x * 16;
  const float vb = Vb[0];

  // issue the (async) copy of e-tile `et` into LDS buffer `buf`
  auto issue_tile = [&](int et, int buf) {
    const float4* gsrc = (const float4*)(enc + (size_t)(b * TENC + et * ETILE) * H_);
    for (int i = t; i < ETILE * (H_ / 4); i += 256) {   // 1024 x float4
      const int r = i >> 5, c4 = i & 31;
#if USE_ASYNC_LDS
      __builtin_amdgcn_global_load_async_to_lds_b128(
          GLOBAL_I4(gsrc + r * (H_ / 4) + c4),
          LDS_I4(&enc_s[buf][r][c4 * 4]), 0, 0);
#else
      ((float4*)&enc_s[buf][r][0])[c4] = gsrc[r * (H_ / 4) + c4];
#endif
    }
  };

  // stage dec_t tile (16 x 128) and V; kick off the first enc tile
  for (int idx = t; idx < 16 * H_; idx += 256)
    dect_s[idx >> 7][idx & 127] =
        dec_t[(size_t)(b * TDEC + d0 + (idx >> 7)) * H_ + (idx & 127)];
  if (t < H_) vw_s[t] = Vw[t];
  issue_tile(0, 0);
#if USE_ASYNC_LDS
  wait_async_zero();
#endif
  __syncthreads();

  // ---- Phase 1: scores (software-pipelined DMA under the tanh loop) -----
  const int d  = t >> 4;              // 0..15 : d-row within tile
  const int el = t & 15;              // 0..15 : e lane
  float* srow = scores + (size_t)(b * TDEC + d0 + d) * TENC;

  for (int et = 0; et < NTILE; ++et) {
    const int cur = et & 1;
    // start DMA of the next tile into the other buffer (freed by the
    // barrier that ended iteration et-1)
    if (et + 1 < NTILE) issue_tile(et + 1, 1 - cur);

    float acc0 = 0.f, acc1 = 0.f;
    #pragma unroll 4
    for (int h = 0; h < H_; ++h) {
      const float dh = dect_s[d][h];  // broadcast within half-wave
      const float v  = vw_s[h];
      acc0 += v * TANH_(dh + enc_s[cur][el     ][h]);
      acc1 += v * TANH_(dh + enc_s[cur][el + 16][h]);
    }
    srow[et * ETILE + el     ] = acc0 + vb;
    srow[et * ETILE + el + 16] = acc1 + vb;

#if USE_ASYNC_LDS
    if (et + 1 < NTILE) wait_async_zero();  // my next-tile DMAs landed
#endif
    __syncthreads();   // all waves: next tile ready, this buffer free
  }

  // ---- Phase 2: softmax (wave w owns rows 2w, 2w+1 — exactly its writes) --
  const int wv   = t >> 5;
  const int lane = t & 31;
  #pragma unroll
  for (int rr = 0; rr < 2; ++rr) {
    float* p = scores + (size_t)(b * TDEC + d0 + wv * 2 + rr) * TENC;
    float m = -__builtin_inff();
    for (int i = lane; i < TENC; i += 32) m = fmaxf(m, p[i]);
    #pragma unroll
    for (int off = 16; off > 0; off >>= 1) m = fmaxf(m, __shfl_xor(m, off, 32));
    float s = 0.f;
    for (int i = lane; i < TENC; i += 32) s += __expf(p[i] - m);
    #pragma unroll
    for (int off = 16; off > 0; off >>= 1) s += __shfl_xor(s, off, 32);
    const float inv = 1.0f / s;
    for (int i = lane; i < TENC; i += 32) p[i] = __expf(p[i] - m) * inv;
  }
  __threadfence();                    // make attn visible across waves
  __syncthreads();

  // ---- Phase 3: context = attn @ enc via fp32 WMMA ------------------------
  // wave wv computes output columns [wv*16, wv*16+16) for the 16 d-rows.
  const int half = lane >> 4;
  const int l16  = lane & 15;
  const int ks   = half * 2;
  const int col  = wv * 16 + l16;
  const float* ap = scores + (size_t)(b * TDEC + d0 + l16) * TENC + ks;
  const float* bp = enc + (size_t)(b * TENC) * H_ + col;

  v8f c;
  #pragma unroll
  for (int r = 0; r < 8; ++r) c[r] = 0.f;

  for (int kk = 0; kk < TENC; kk += 4) {
    v2f a, bb;
    a.x  = ap[kk];                 a.y  = ap[kk + 1];
    bb.x = bp[(kk + ks) * H_];     bb.y = bp[(kk + ks + 1) * H_];
    c = __builtin_amdgcn_wmma_f32_16x16x4_f32(false, a, false, bb,
                                              (short)0, c, false, false);
  }

  const int mbase = d0 + half * 8;
  #pragma unroll
  for (int r = 0; r < 8; ++r)
    out[(size_t)(b * TDEC + mbase + r) * H_ + col] = c[r];
}

// ---------------------------------------------------------------------------
extern "C" void kernel_launch(void* const* d_in, const int* in_sizes, int n_in,
                              void* d_out, int out_size, void* d_ws, size_t ws_size,
                              hipStream_t stream) {
  (void)in_sizes; (void)n_in; (void)out_size; (void)ws_size;
  const float* dec = (const float*)d_in[0];  // (B, TDEC, H)
  const float* enc = (const float*)d_in[1];  // (B, TENC, H)
  const float* Ww  = (const float*)d_in[2];  // (H, H)
  const float* Wb  = (const float*)d_in[3];  // (H)
  const float* Vw  = (const float*)d_in[4];  // (H)
  const float* Vb  = (const float*)d_in[5];  // scalar
  float* out = (float*)d_out;

  float* dec_t  = (float*)d_ws;                          // 1 MB
  float* scores = dec_t + (size_t)B_ * TDEC * H_;        // 8 MB

  dim3 gA(B_ * TDEC / 16, H_ / 16);                      // 128 x 8 tiles
  dec_transform_wmma<<<gA, 32, 0, stream>>>(dec, Ww, Wb, dec_t);

  dim3 gB(TDEC / 16, B_);                                // 32 x 4 blocks
  attn_fused<<<gB, 256, 0, stream>>>(enc, Vw, Vb, dec_t, scores, out);
}